// Local_Glbal_LC_1168231104604
// MI455X (gfx1250) — compile-verified
//
#include <hip/hip_runtime.h>
#include <hip/hip_bf16.h>
#include <math.h>

// Probe only in the device pass: the builtin is declared for gfx1250 but not x86.
#if defined(__HIP_DEVICE_COMPILE__)
#if !__has_builtin(__builtin_amdgcn_wmma_f32_16x16x4_f32)
#error "wmma f32 16x16x4 builtin not available on this device toolchain"
#endif
#endif

typedef float v2f __attribute__((ext_vector_type(2)));
typedef float v8f __attribute__((ext_vector_type(8)));

// ---------------------------------------------------------------------------
// Graph-norm kernels
// ---------------------------------------------------------------------------
__global__ void deg_kernel(const int* __restrict__ dst, float* __restrict__ deg, int E) {
    int e = blockIdx.x * blockDim.x + threadIdx.x;
    if (e < E) atomicAdd(&deg[dst[e]], 1.0f);
}

__global__ void dinv_kernel(float* __restrict__ deg, int N) {
    int i = blockIdx.x * blockDim.x + threadIdx.x;
    if (i < N) deg[i] = rsqrtf(deg[i] + 1.0f);   // +1 = self loop; always > 0
}

__global__ void norm_kernel(const int* __restrict__ src, const int* __restrict__ dst,
                            const float* __restrict__ dinv, float* __restrict__ norm, int E) {
    int e = blockIdx.x * blockDim.x + threadIdx.x;
    if (e < E) norm[e] = dinv[src[e]] * dinv[dst[e]];
}

// ---------------------------------------------------------------------------
// WMMA f32 GEMM:  C[M,KOUT] = A[M,KIN] @ W[KIN,KOUT] (+bias) (+sigmoid)
// One wave computes one 16x16 output tile; K stepped by 4 via
// V_WMMA_F32_16X16X4_F32.  Requires M % 16 == 0 (true here: 50000 = 3125*16).
// SPLIT: A columns k<128 read from A1 (row width 128), k>=128 from A2 (width 64)
// to implement concat([h, hl]) without materializing it.
// ---------------------------------------------------------------------------
template <int KIN, int KOUT, bool SPLIT, bool BIAS, bool SIGMOID>
__global__ void __launch_bounds__(128)
gemm_wmma_f32(const float* __restrict__ A1, const float* __restrict__ A2,
              const float* __restrict__ W, const float* __restrict__ bias,
              float* __restrict__ C, int M) {
    const int tilesN = KOUT / 16;
    const int wave = (blockIdx.x * blockDim.x + threadIdx.x) >> 5;
    const int totalTiles = (M / 16) * tilesN;
    if (wave >= totalTiles) return;                // whole-wave uniform exit
    const int lane = threadIdx.x & 31;
    const int g = lane >> 4;                       // half-wave: 0 or 1
    const int l = lane & 15;
    const int tm = wave / tilesN;
    const int tn = wave % tilesN;
    const int m = tm * 16 + l;                     // A row for this lane
    const int n = tn * 16 + l;                     // B/C column for this lane

    v8f acc = {};
#pragma unroll
    for (int k0 = 0; k0 < KIN; k0 += 4) {
        const int ka = k0 + 2 * g;                 // lanes 0-15: K=k0,k0+1; 16-31: k0+2,k0+3
        v2f a, b;
        if (SPLIT && ka >= 128) {
            const float* pa = A2 + (size_t)m * 64 + (ka - 128);
            a[0] = pa[0]; a[1] = pa[1];
        } else {
            const int lda = SPLIT ? 128 : KIN;
            const float* pa = A1 + (size_t)m * lda + ka;
            a[0] = pa[0]; a[1] = pa[1];
        }
        b[0] = W[(size_t)ka * KOUT + n];
        b[1] = W[(size_t)(ka + 1) * KOUT + n];
        // (neg_a, A, neg_b, B, c_mod, C, reuse_a, reuse_b)
        acc = __builtin_amdgcn_wmma_f32_16x16x4_f32(false, a, false, b,
                                                    (short)0, acc, false, false);
    }

#pragma unroll
    for (int v = 0; v < 8; ++v) {
        const int row = tm * 16 + g * 8 + v;       // C layout: vgpr v, half g -> M = v + 8g
        float val = acc[v];
        if (BIAS) val += bias[n];
        if (SIGMOID) val = 1.0f / (1.0f + expf(-val));
        C[(size_t)row * KOUT + n] = val;
    }
}

// ---------------------------------------------------------------------------
// Edge aggregation: out[dst] += tmp[src] * norm  (float atomics, resolved in L2)
// One thread per (edge, 4-channel chunk); consecutive threads cover consecutive
// channels of the same edge -> coalesced 128B/edge row accesses.
// ---------------------------------------------------------------------------
template <int DIM>
__global__ void scatter_agg(const float* __restrict__ tmp, const int* __restrict__ src,
                            const int* __restrict__ dst, const float* __restrict__ norm,
                            float* __restrict__ out, int E) {
    const int C4 = DIM / 4;
    long long t = (long long)blockIdx.x * blockDim.x + threadIdx.x;
    int e = (int)(t / C4);
    if (e >= E) return;
    int c = (int)(t % C4) * 4;
    int s = src[e], d = dst[e];
    float w = norm[e];
    const float4 v = *(const float4*)(tmp + (size_t)s * DIM + c);
    float* o = out + (size_t)d * DIM + c;
    atomicAdd(o + 0, v.x * w);
    atomicAdd(o + 1, v.y * w);
    atomicAdd(o + 2, v.z * w);
    atomicAdd(o + 3, v.w * w);
}

// ---------------------------------------------------------------------------
// Epilogue: fold self-loop contribution (tmp[i]*dinv[i]^2), bias, ReLU
// ---------------------------------------------------------------------------
template <int DIM, bool RELU>
__global__ void epilogue(float* __restrict__ out, const float* __restrict__ tmp,
                         const float* __restrict__ dinv, const float* __restrict__ bias,
                         int N) {
    long long t = (long long)blockIdx.x * blockDim.x + threadIdx.x;
    if (t >= (long long)N * DIM) return;
    int i = (int)(t / DIM);
    int c = (int)(t % DIM);
    float di = dinv[i];
    float v = out[t] + tmp[t] * di * di + bias[c];
    out[t] = RELU ? fmaxf(v, 0.0f) : v;
}

// ---------------------------------------------------------------------------
static inline int cdivll(long long a, int b) { return (int)((a + b - 1) / b); }

extern "C" void kernel_launch(void* const* d_in, const int* in_sizes, int n_in,
                              void* d_out, int out_size, void* d_ws, size_t ws_size,
                              hipStream_t stream) {
    (void)n_in; (void)out_size; (void)ws_size;

    const float* x        = (const float*)d_in[0];   // [N,128]
    const float* y        = (const float*)d_in[1];   // [N,64]
    const int*   ei       = (const int*)d_in[2];     // [2,E]
    const float* LC       = (const float*)d_in[3];   // [64,64]
    const float* gcn_w    = (const float*)d_in[4];   // [2,128,128]
    const float* gcn_b    = (const float*)d_in[5];   // [2,128]
    const float* label_w  = (const float*)d_in[6];   // [10,64,64]
    const float* label_b  = (const float*)d_in[7];   // [10,64]
    const float* fusion_w = (const float*)d_in[8];   // [192,64]
    const float* fusion_b = (const float*)d_in[9];   // [64]

    const int N = in_sizes[0] / 128;
    const int E = in_sizes[2] / 2;
    const int* src = ei;
    const int* dst = ei + E;

    // Workspace layout (floats): dinv | norm | tmp(N*128, also fusion out) | hA(N*128) | lA(N*64)
    float* dinv = (float*)d_ws;
    float* norm = dinv + N;
    float* tmp  = norm + E;
    float* hA   = tmp + (size_t)N * 128;
    float* lA   = hA + (size_t)N * 128;
    float* out  = (float*)d_out;

    const int TB = 256;

    // --- symmetric normalization ---
    (void)hipMemsetAsync(dinv, 0, (size_t)N * sizeof(float), stream);
    deg_kernel<<<cdivll(E, TB), TB, 0, stream>>>(dst, dinv, E);
    dinv_kernel<<<cdivll(N, TB), TB, 0, stream>>>(dinv, N);
    norm_kernel<<<cdivll(E, TB), TB, 0, stream>>>(src, dst, dinv, norm, E);

    const int waves128 = (N / 16) * (128 / 16);   // tiles for KOUT=128
    const int waves64  = (N / 16) * (64 / 16);    // tiles for KOUT=64
    const int gb128 = (waves128 * 32 + 127) / 128;
    const int gb64  = (waves64 * 32 + 127) / 128;

    // --- GCN layer 0: x -> hA (relu) ---
    gemm_wmma_f32<128, 128, false, false, false>
        <<<gb128, 128, 0, stream>>>(x, nullptr, gcn_w, nullptr, tmp, N);
    (void)hipMemsetAsync(hA, 0, (size_t)N * 128 * sizeof(float), stream);
    scatter_agg<128><<<cdivll((long long)E * 32, TB), TB, 0, stream>>>(tmp, src, dst, norm, hA, E);
    epilogue<128, true><<<cdivll((long long)N * 128, TB), TB, 0, stream>>>(hA, tmp, dinv, gcn_b, N);

    // --- GCN layer 1: hA -> hA (no relu) ---
    gemm_wmma_f32<128, 128, false, false, false>
        <<<gb128, 128, 0, stream>>>(hA, nullptr, gcn_w + 128 * 128, nullptr, tmp, N);
    (void)hipMemsetAsync(hA, 0, (size_t)N * 128 * sizeof(float), stream);
    scatter_agg<128><<<cdivll((long long)E * 32, TB), TB, 0, stream>>>(tmp, src, dst, norm, hA, E);
    epilogue<128, false><<<cdivll((long long)N * 128, TB), TB, 0, stream>>>(hA, tmp, dinv, gcn_b + 128, N);

    // --- 10 label-propagation layers: y -> lA ---
    for (int j = 0; j < 10; ++j) {
        const float* in = (j == 0) ? y : lA;
        gemm_wmma_f32<64, 64, false, false, false>
            <<<gb64, 128, 0, stream>>>(in, nullptr, label_w + (size_t)j * 64 * 64, nullptr, tmp, N);
        (void)hipMemsetAsync(lA, 0, (size_t)N * 64 * sizeof(float), stream);
        scatter_agg<64><<<cdivll((long long)E * 16, TB), TB, 0, stream>>>(tmp, src, dst, norm, lA, E);
        if (j < 9)
            epilogue<64, true><<<cdivll((long long)N * 64, TB), TB, 0, stream>>>(lA, tmp, dinv, label_b + j * 64, N);
        else
            epilogue<64, false><<<cdivll((long long)N * 64, TB), TB, 0, stream>>>(lA, tmp, dinv, label_b + j * 64, N);
    }

    // --- fusion: Z = concat(hA, lA) @ fusion_w + fusion_b   (Z reuses tmp) ---
    gemm_wmma_f32<192, 64, true, true, false>
        <<<gb64, 128, 0, stream>>>(hA, lA, fusion_w, fusion_b, tmp, N);

    // --- final: out = sigmoid(Z @ LC) ---
    gemm_wmma_f32<64, 64, false, false, true>
        <<<gb64, 128, 0, stream>>>(tmp, nullptr, LC, nullptr, out, N);
}